// ContrastiveLoss_7035156431246
// MI455X (gfx1250) — compile-verified
//
#include <hip/hip_runtime.h>
#include <hip/hip_bf16.h>

// ---------------- problem constants (from setup_inputs) ----------------
#define BB   32          // videos
#define CC   256         // channels
#define NN   4096        // N*N (N=64)
#define SS   128         // sentences
#define MM   256         // moments
#define PP   2080        // valid proposals (upper-tri of 64x64)
#define TN   32          // z-tile width for the big fused GEMM
#define CS   (CC + 2)    // padded LDS row stride (even: keeps 8B align, spreads banks)
#define INV_T 10.0f      // 1/T_V == 1/T_Q == 10
#define NEG_IOU 0.5f

typedef float v2f __attribute__((ext_vector_type(2)));
typedef float v8f __attribute__((ext_vector_type(8)));

// D = A(16x4 f32) * B(4x16 f32) + C(16x16 f32)   — CDNA5 fp32 WMMA
static __device__ inline v8f wmma_f32_16x16x4(v2f a, v2f b, v8f c) {
  return __builtin_amdgcn_wmma_f32_16x16x4_f32(
      /*neg_a=*/false, a, /*neg_b=*/false, b,
      /*c_mod=*/(short)0, c, /*reuse_a=*/false, /*reuse_b=*/false);
}

// sum across the 16 lanes that share an output row (lanes 0-15 / 16-31 separately)
static __device__ inline float half_wave_sum(float e) {
#pragma unroll
  for (int off = 1; off < 16; off <<= 1) e += __shfl_xor(e, off, 32);
  return e;
}

// ---------------- k0: normalize sentence features -> sf [S,C] ----------------
__global__ __launch_bounds__(256) void k0_sents(const float* __restrict__ sents,
                                                float* __restrict__ sf) {
  __shared__ float red[256];
  const int s = blockIdx.x, c = threadIdx.x;
  float v = sents[s * CC + c];
  red[c] = v * v;
  __syncthreads();
  for (int st = 128; st > 0; st >>= 1) {
    if (c < st) red[c] += red[c + st];
    __syncthreads();
  }
  float rn = 1.0f / fmaxf(sqrtf(red[0]), 1e-12f);
  sf[s * CC + c] = v * rn;
}

// -------- k1: reciprocal norms of gathered video columns -> rnorm [B*P] ------
__global__ __launch_bounds__(256) void k1_rnorm(const float* __restrict__ vfeat,
                                                const int* __restrict__ midx,
                                                float* __restrict__ rnorm) {
  const int b = blockIdx.y;
  const int p = blockIdx.x * 256 + threadIdx.x;
  if (p >= PP) return;
  const int g = midx[p];
  const float* base = vfeat + ((size_t)b * CC) * NN;
  float ss = 0.0f;
#pragma unroll 4
  for (int c = 0; c < CC; ++c) {
    float v = base[((size_t)c << 12) + g];   // coalesced across p (contiguous runs)
    ss += v * v;
  }
  rnorm[b * PP + p] = 1.0f / fmaxf(sqrtf(ss), 1e-12f);
}

// -------- k1b: iou2d_f > NEG_IOU byte mask -> ioub [S*P] ----------------------
__global__ __launch_bounds__(256) void k1b_ioub(const float* __restrict__ iou2d,
                                                const int* __restrict__ midx,
                                                unsigned char* __restrict__ ioub) {
  int i = blockIdx.x * 256 + threadIdx.x;
  if (i >= SS * PP) return;
  int s = i / PP, p = i - s * PP;
  ioub[i] = (iou2d[s * NN + midx[p]] > NEG_IOU) ? 1 : 0;
}

// -------- k2: fused big GEMM sf[S,C] x vf[z,C]^T with masked exp row-sums -----
// one workgroup per 32-column z-tile (2080 wgs); 8 waves cover all 128 s-rows.
__global__ __launch_bounds__(256) void k2_query(const float* __restrict__ vfeat,
                                                const float* __restrict__ sf,
                                                const float* __restrict__ rnorm,
                                                const unsigned char* __restrict__ ioub,
                                                const int* __restrict__ midx,
                                                const int* __restrict__ s2v,
                                                float* __restrict__ qsum) {
  __shared__ __align__(16) float Bt[TN * CS];  // vf tile, [z][c] padded, ~32.3 KB
  __shared__ int   gi[TN];
  __shared__ float rns[TN];
  __shared__ float qs[SS];
  const int tid = threadIdx.x;
  const int z0 = blockIdx.x * TN;
  const int b  = z0 / PP;
  const int p0 = z0 - b * PP;     // tile never crosses a video boundary (P%32==0)

  if (tid < TN) { gi[tid] = midx[p0 + tid]; rns[tid] = rnorm[z0 + tid]; }
  for (int i = tid; i < SS; i += 256) qs[i] = 0.0f;
  __syncthreads();

  const float* vb = vfeat + ((size_t)b * CC) * NN;
  for (int i = tid; i < CC * TN; i += 256) {
    int zl = i & (TN - 1);        // fast index -> coalesced global gather
    int c  = i >> 5;
    Bt[zl * CS + c] = vb[((size_t)c << 12) + gi[zl]] * rns[zl];  // normalized on the fly
  }
  __syncthreads();

  const int w  = tid >> 5;        // wave -> row band
  const int l  = tid & 31;
  const int lr = l & 15;
  const int hi = l >> 4;
  const int r0 = w * 16;

  v8f acc[2] = {};
  for (int k0 = 0; k0 < CC; k0 += 4) {
    const int kk = k0 + 2 * hi;                            // even -> 8B aligned
    v2f a  = *(const v2f*)(sf + (r0 + lr) * CC + kk);      // A: sf rows (L2-hot)
    v2f b0 = *(const v2f*)(Bt + lr * CS + kk);             // one ds_load_b64 each
    v2f b1 = *(const v2f*)(Bt + (16 + lr) * CS + kk);
    acc[0] = wmma_f32_16x16x4(a, b0, acc[0]);
    acc[1] = wmma_f32_16x16x4(a, b1, acc[1]);
  }

  // epilogue: branchless mask + exp; butterfly-reduce the 16 lanes sharing a
  // row, then a single LDS atomic per row-group
  for (int t = 0; t < 2; ++t) {
    const int col = t * 16 + lr;
    const int p   = p0 + col;
#pragma unroll
    for (int i = 0; i < 8; ++i) {
      const int s = r0 + hi * 8 + i;
      const int keep = 1 - ((s2v[s] == b) & (int)ioub[s * PP + p]);  // no short-circuit
      float e = __expf(acc[t][i] * INV_T) * (float)keep;
      e = half_wave_sum(e);
      if (lr == 0) atomicAdd(&qs[s], e);
    }
  }
  __syncthreads();
  for (int i = tid; i < SS; i += 256) atomicAdd(&qsum[i], qs[i]);
}

// -------- k3: top-1 proposal per moment + gather normalized tv [M,C] ---------
__global__ __launch_bounds__(256) void k3_top1(const float* __restrict__ vfeat,
                                               const float* __restrict__ iou2ds,
                                               const int* __restrict__ midx,
                                               const float* __restrict__ rnorm,
                                               const int* __restrict__ s2v,
                                               const int* __restrict__ m2s,
                                               float* __restrict__ tv) {
  __shared__ float bv[256];
  __shared__ int   bp[256];
  const int m = blockIdx.x, tid = threadIdx.x;
  float best = -1e30f; int bidx = PP;
  for (int p = tid; p < PP; p += 256) {
    float v = iou2ds[m * NN + midx[p]];
    if (v > best) { best = v; bidx = p; }   // keeps lowest p on ties per thread
  }
  bv[tid] = best; bp[tid] = bidx;
  __syncthreads();
  for (int st = 128; st > 0; st >>= 1) {
    if (tid < st) {
      float v2 = bv[tid + st]; int p2 = bp[tid + st];
      if (v2 > bv[tid] || (v2 == bv[tid] && p2 < bp[tid])) { bv[tid] = v2; bp[tid] = p2; }
    }
    __syncthreads();
  }
  const int pbest = bp[0];
  const int vid   = s2v[m2s[m]];
  const int g     = midx[pbest];
  const float rn  = rnorm[vid * PP + pbest];
  for (int c = tid; c < CC; c += 256)
    tv[m * CC + c] = vfeat[((size_t)(vid * CC + c) << 12) + g] * rn;
}

// -------- k4: tv[M,C] x sf[S,C]^T (256x128, K=256) + pos + masked exp sums ---
// 16 workgroups; wg g owns m-rows [16g,16g+16); wave w owns s-tile w.
__global__ __launch_bounds__(256) void k4_video(const float* __restrict__ tv,
                                                const float* __restrict__ sf,
                                                const int* __restrict__ m2s,
                                                float* __restrict__ vsum,
                                                float* __restrict__ pos) {
  __shared__ float vs[16];
  const int tid = threadIdx.x;
  const int w = tid >> 5, l = tid & 31, lr = l & 15, hi = l >> 4;
  const int m0 = blockIdx.x * 16;
  const int s0 = w * 16;
  if (tid < 16) vs[tid] = 0.0f;
  __syncthreads();

  v8f acc = {};
  for (int k0 = 0; k0 < CC; k0 += 4) {
    const int kk = k0 + 2 * hi;
    v2f a = *(const v2f*)(tv + (m0 + lr) * CC + kk);
    v2f b = *(const v2f*)(sf + (s0 + lr) * CC + kk);
    acc = wmma_f32_16x16x4(a, b, acc);
  }
#pragma unroll
  for (int i = 0; i < 8; ++i) {
    const int m = m0 + hi * 8 + i;
    const int s = s0 + lr;
    const float v = acc[i];
    const int sm = m2s[m];
    if (s == sm) pos[m] = v;                       // exactly one writer per m
    float e = __expf(v * INV_T) * (float)(s != sm);
    e = half_wave_sum(e);
    if (lr == 0) atomicAdd(&vs[m - m0], e);
  }
  __syncthreads();
  if (tid < 16) vsum[m0 + tid] = vs[tid];          // rows partitioned: plain store
}

// -------- k5: final log-CE reduction -> 3 scalars ----------------------------
__global__ __launch_bounds__(256) void k5_final(const float* __restrict__ pos,
                                                const float* __restrict__ vsum,
                                                const float* __restrict__ qsum,
                                                const int* __restrict__ m2s,
                                                float* __restrict__ out) {
  __shared__ float rv[256], rq[256];
  const int m = threadIdx.x;
  const float p  = pos[m] * INV_T;
  const float pe = __expf(p);
  rv[m] = __logf(pe + vsum[m]) - p;
  rq[m] = __logf(pe + qsum[m2s[m]]) - p;
  __syncthreads();
  for (int st = 128; st > 0; st >>= 1) {
    if (m < st) { rv[m] += rv[m + st]; rq[m] += rq[m + st]; }
    __syncthreads();
  }
  if (m == 0) {
    out[0] = rv[0] * (1.0f / MM);
    out[1] = rq[0] * (1.0f / MM);
    out[2] = 0.0f;
  }
}

// ---------------- launcher ----------------
extern "C" void kernel_launch(void* const* d_in, const int* in_sizes, int n_in,
                              void* d_out, int out_size, void* d_ws, size_t ws_size,
                              hipStream_t stream) {
  const float* vfeat  = (const float*)d_in[0];   // [B,C,N,N]
  const float* sents  = (const float*)d_in[1];   // [S,C]
  const float* iou2d  = (const float*)d_in[2];   // [S,N,N]
  const float* iou2ds = (const float*)d_in[3];   // [M,N,N]
  const int*   midx   = (const int*)d_in[4];     // [P]
  const int*   s2v    = (const int*)d_in[5];     // [S]
  const int*   m2s    = (const int*)d_in[6];     // [M]
  float* out = (float*)d_out;

  char* ws = (char*)d_ws;
  float* sf    = (float*)(ws + 0);               // S*C*4      = 131072
  float* rnorm = (float*)(ws + 131072);          // B*P*4      = 266240
  float* tv    = (float*)(ws + 397312);          // M*C*4      = 262144
  float* qsum  = (float*)(ws + 659456);          // S*4        = 512
  float* vsum  = (float*)(ws + 659968);          // M*4        = 1024
  float* pos   = (float*)(ws + 660992);          // M*4        = 1024
  unsigned char* ioub = (unsigned char*)(ws + 662016);  // S*P = 266240

  hipMemsetAsync(qsum, 0, SS * sizeof(float), stream);

  k0_sents <<<SS, 256, 0, stream>>>(sents, sf);
  k1_rnorm <<<dim3((PP + 255) / 256, BB), 256, 0, stream>>>(vfeat, midx, rnorm);
  k1b_ioub <<<(SS * PP + 255) / 256, 256, 0, stream>>>(iou2d, midx, ioub);
  k2_query <<<(BB * PP) / TN, 256, 0, stream>>>(vfeat, sf, rnorm, ioub, midx, s2v, qsum);
  k3_top1  <<<MM, 256, 0, stream>>>(vfeat, iou2ds, midx, rnorm, s2v, m2s, tv);
  k4_video <<<MM / 16, 256, 0, stream>>>(tv, sf, m2s, vsum, pos);
  k5_final <<<1, 256, 0, stream>>>(pos, vsum, qsum, m2s, out);
}